// CPDecode_66709432041912
// MI455X (gfx1250) — compile-verified
//
#include <hip/hip_runtime.h>
#include <math.h>

// ---------------- problem constants (from reference) ----------------
#define L_   5
#define H_   8
#define KVH_ 8
#define D_   128
#define HID_ 1024
#define FF_  4096
#define V_   2048
#define NCB_ 15
#define S_   4096
#define LEN_ (S_ + 1)        // 4097
#define SP_  4112            // padded per-head score stride (multiple of 16)

typedef __attribute__((ext_vector_type(2))) float v2f;
typedef __attribute__((ext_vector_type(8))) float v8f;

// ---------------- cache concat copy: old [2L,KVH,S,D] -> new [2L,KVH,S+1,D] ----------------
__global__ void ck_copy_cache(const float4* __restrict__ src, float4* __restrict__ dst) {
    const size_t chunk = (size_t)S_ * D_ / 4;          // per (slab,head) f4 count = 131072
    const size_t dchunk = (size_t)LEN_ * D_ / 4;       // dst stride = 131104
    const size_t total = (size_t)2 * L_ * KVH_ * chunk;
    size_t stride = (size_t)gridDim.x * blockDim.x;
    for (size_t i = (size_t)blockIdx.x * blockDim.x + threadIdx.x; i < total; i += stride) {
        size_t sh = i / chunk, within = i - sh * chunk;
        dst[sh * dchunk + within] = src[i];
    }
}

// ---------------- RMSNorm: out = x * rsqrt(mean(x^2)+eps) * w ----------------
__global__ void ck_rmsnorm(const float* __restrict__ x, const float* __restrict__ w,
                           float* __restrict__ out, int n) {
    __shared__ float red[256];
    float s = 0.f;
    for (int i = threadIdx.x; i < n; i += 256) { float t = x[i]; s += t * t; }
    red[threadIdx.x] = s; __syncthreads();
    for (int st = 128; st; st >>= 1) {
        if ((int)threadIdx.x < st) red[threadIdx.x] += red[threadIdx.x + st];
        __syncthreads();
    }
    float rs = rsqrtf(red[0] / (float)n + 1e-6f);
    for (int i = threadIdx.x; i < n; i += 256) out[i] = x[i] * rs * w[i];
}

// ---------------- generic matvec: y[r] = (res?res[r]:0) + dot(W[r,:], x) ----------------
// one wave (32 lanes) per output row, float4 streaming loads; 8 rows per 256-thread block
__global__ void ck_matvec(const float* __restrict__ W, const float* __restrict__ x,
                          float* __restrict__ y, const float* __restrict__ res,
                          int rows, int cols) {
    int wave = threadIdx.x >> 5, lane = threadIdx.x & 31;
    int r = blockIdx.x * 8 + wave;
    if (r >= rows) return;
    const float* wr = W + (size_t)r * cols;
    float acc = 0.f;
    for (int c = lane * 4; c < cols; c += 128) {
        float4 wv = *(const float4*)(wr + c);
        float4 xv = *(const float4*)(x + c);
        acc += wv.x * xv.x + wv.y * xv.y + wv.z * xv.z + wv.w * xv.w;
    }
    for (int off = 16; off; off >>= 1) acc += __shfl_down(acc, off, 32);
    if (lane == 0) y[r] = (res ? res[r] : 0.f) + acc;
}

// ---------------- gated FF rows: act[r] = silu(Wg[r]·x) * (Wu[r]·x) ----------------
__global__ void ck_gated(const float* __restrict__ Wg, const float* __restrict__ Wu,
                         const float* __restrict__ x, float* __restrict__ act,
                         int rows, int cols) {
    int wave = threadIdx.x >> 5, lane = threadIdx.x & 31;
    int r = blockIdx.x * 8 + wave;
    if (r >= rows) return;
    const float* gr = Wg + (size_t)r * cols;
    const float* ur = Wu + (size_t)r * cols;
    float ga = 0.f, ua = 0.f;
    for (int c = lane * 4; c < cols; c += 128) {
        float4 xv = *(const float4*)(x + c);
        float4 gv = *(const float4*)(gr + c);
        float4 uv = *(const float4*)(ur + c);
        ga += gv.x * xv.x + gv.y * xv.y + gv.z * xv.z + gv.w * xv.w;
        ua += uv.x * xv.x + uv.y * xv.y + uv.z * xv.z + uv.w * xv.w;
    }
    for (int off = 16; off; off >>= 1) {
        ga += __shfl_down(ga, off, 32);
        ua += __shfl_down(ua, off, 32);
    }
    if (lane == 0) {
        float sg = ga / (1.f + expf(-ga));
        act[r] = sg * ua;
    }
}

// ---------------- RoPE on q,k (in place) + append roped k and v to new cache ----------------
// one thread per rotation pair (head, j) with j in [0,64)
__global__ void ck_rope_append(float* __restrict__ q, float* __restrict__ k,
                               const float* __restrict__ v, float* __restrict__ onc,
                               int layer, const int* __restrict__ posp) {
    int tid = blockIdx.x * 256 + threadIdx.x;      // 0..511
    if (tid >= H_ * (D_ / 2)) return;
    int head = tid >> 6;
    int j = tid & 63;
    float p = (float)posp[0];
    float inv = powf(10000.f, -((float)(2 * j)) / (float)D_);
    float ang = p * inv;
    float c = cosf(ang), s = sinf(ang);
    int b = head * D_;
    float q1 = q[b + j], q2 = q[b + j + 64];
    q[b + j]      = q1 * c - q2 * s;
    q[b + j + 64] = q2 * c + q1 * s;
    float k1 = k[b + j], k2 = k[b + j + 64];
    float ko1 = k1 * c - k2 * s;
    float ko2 = k2 * c + k1 * s;
    k[b + j] = ko1; k[b + j + 64] = ko2;
    size_t kd = ((size_t)(2 * layer) * KVH_ + head) * (size_t)LEN_ * D_ + (size_t)S_ * D_;
    size_t vd = ((size_t)(2 * layer + 1) * KVH_ + head) * (size_t)LEN_ * D_ + (size_t)S_ * D_;
    onc[kd + j] = ko1;        onc[kd + j + 64] = ko2;
    onc[vd + j] = v[b + j];   onc[vd + j + 64] = v[b + j + 64];
}

// ---------------- attention scores via fp32 WMMA (V_WMMA_F32_16X16X4_F32) ----------------
// One wave per 16-row tile of K-cache rows; accumulate over D=128 in steps of 4.
// A (16x4 f32): lanes 0-15 hold M=lane, K=0,1 in vgpr0/1; lanes 16-31 hold M=lane-16,
// K=2,3. B (4x16) mirrors with (K,N) swapped; we broadcast the q chunk across all N
// columns so any column of D carries the scores. No masking in the loop: D rows beyond
// LEN_ accumulate garbage but are never written (write guard below); their lanes point
// at the always-valid knew row so all loads are safe and branch-free.
__global__ __launch_bounds__(32)
void ck_attn_scores(const float* __restrict__ Kc,   // layer K base [KVH][S][D]
                    const float* __restrict__ knew, // [KVH][D] (already roped)
                    const float* __restrict__ q,    // [H][D]   (already roped)
                    float* __restrict__ scores) {   // [H][SP_]
    int head = blockIdx.y;
    int base = blockIdx.x * 16;
    int lane = threadIdx.x;                  // 0..31
    int m = lane & 15;
    int row = base + m;
    int kofs = (lane >= 16) ? 2 : 0;
    // per-lane row pointer; out-of-range rows read the (valid) new-k row harmlessly
    const float* rp = (row < S_) ? (Kc + ((size_t)head * S_ + row) * D_)
                                 : (knew + head * D_);
    const float* ap = rp + kofs;             // A fragment stream (8B aligned)
    const float* bp = q + head * D_ + kofs;  // B fragment stream (8B aligned)
    v8f c = {0.f, 0.f, 0.f, 0.f, 0.f, 0.f, 0.f, 0.f};
#pragma unroll 4
    for (int kk = 0; kk < D_; kk += 4) {
        v2f a = *(const v2f*)(ap + kk);      // global_load_b64
        v2f b = *(const v2f*)(bp + kk);      // global_load_b64 (cached, tiny)
        c = __builtin_amdgcn_wmma_f32_16x16x4_f32(
                /*neg_a=*/false, a, /*neg_b=*/false, b,
                /*c_mod=*/(short)0, c, /*reuse_a=*/false, /*reuse_b=*/false);
    }
    const float scale = 0.08838834764831845f;   // 1/sqrt(128)
    // C/D layout: vgpr r -> M=r (lanes 0-15) / M=8+r (lanes 16-31), N = lane%16. Col 0:
    if (lane == 0) {
        for (int r2 = 0; r2 < 8; ++r2) {
            int rr = base + r2;
            if (rr < LEN_) scores[head * SP_ + rr] = c[r2] * scale;
        }
    }
    if (lane == 16) {
        for (int r2 = 0; r2 < 8; ++r2) {
            int rr = base + 8 + r2;
            if (rr < LEN_) scores[head * SP_ + rr] = c[r2] * scale;
        }
    }
}

// ---------------- softmax over len entries, per head, in place ----------------
__global__ void ck_softmax(float* __restrict__ scores, int len) {
    int head = blockIdx.x;
    float* s = scores + head * SP_;
    __shared__ float red[256];
    float m = -INFINITY;
    for (int i = threadIdx.x; i < len; i += 256) m = fmaxf(m, s[i]);
    red[threadIdx.x] = m; __syncthreads();
    for (int st = 128; st; st >>= 1) {
        if ((int)threadIdx.x < st) red[threadIdx.x] = fmaxf(red[threadIdx.x], red[threadIdx.x + st]);
        __syncthreads();
    }
    m = red[0]; __syncthreads();
    float sum = 0.f;
    for (int i = threadIdx.x; i < len; i += 256) { float e = expf(s[i] - m); s[i] = e; sum += e; }
    red[threadIdx.x] = sum; __syncthreads();
    for (int st = 128; st; st >>= 1) {
        if ((int)threadIdx.x < st) red[threadIdx.x] += red[threadIdx.x + st];
        __syncthreads();
    }
    float invv = 1.f / red[0];
    for (int i = threadIdx.x; i < len; i += 256) s[i] *= invv;
}

// ---------------- o[h,d] = sum_s p[h,s] * V[h,s,d] (incl. new token) ----------------
__global__ void ck_attn_av(const float* __restrict__ Vc, const float* __restrict__ vnew,
                           const float* __restrict__ p, float* __restrict__ o) {
    int head = blockIdx.x;
    int d = threadIdx.x;        // 0..127
    int sl = threadIdx.y;       // 0..3
    const float* Vh = Vc + (size_t)head * S_ * D_;
    const float* ph = p + head * SP_;
    float acc = 0.f;
    for (int s = sl; s < S_; s += 4) acc += ph[s] * Vh[(size_t)s * D_ + d];
    if (sl == 0) acc += ph[S_] * vnew[head * D_ + d];
    __shared__ float red[4][128];
    red[sl][d] = acc; __syncthreads();
    if (sl == 0) o[head * D_ + d] = red[0][d] + red[1][d] + red[2][d] + red[3][d];
}

// ---------------- host-side orchestration ----------------
extern "C" void kernel_launch(void* const* d_in, const int* in_sizes, int n_in,
                              void* d_out, int out_size, void* d_ws, size_t ws_size,
                              hipStream_t stream) {
    const float* in_embed = (const float*)d_in[0];
    const float* kvc      = (const float*)d_in[1];
    const int*   posp     = (const int*)d_in[2];
    const float* ln1      = (const float*)d_in[3];
    const float* Wq       = (const float*)d_in[4];
    const float* Wk       = (const float*)d_in[5];
    const float* Wv       = (const float*)d_in[6];
    const float* Wo       = (const float*)d_in[7];
    const float* ln2      = (const float*)d_in[8];
    const float* Wg       = (const float*)d_in[9];
    const float* Wu       = (const float*)d_in[10];
    const float* Wd       = (const float*)d_in[11];
    const float* normw    = (const float*)d_in[12];
    const float* lmw      = (const float*)d_in[13];

    float* out    = (float*)d_out;
    float* logits = out;
    float* onc    = out + (size_t)NCB_ * V_;        // new cache region

    float* ws  = (float*)d_ws;
    float* x   = ws;            // [1024]
    float* h   = ws + 1024;     // [1024]
    float* q   = ws + 2048;     // [1024]
    float* k   = ws + 3072;     // [1024]
    float* v   = ws + 4096;     // [1024]
    float* o   = ws + 5120;     // [1024]
    float* act = ws + 6144;     // [4096]
    float* sc  = ws + 10240;    // [H * SP_] = 32896

    // old-cache -> new-cache bulk copy (exact grid: 40960*256 f4 slots)
    ck_copy_cache<<<40960, 256, 0, stream>>>((const float4*)kvc, (float4*)onc);
    // x = input_embed
    hipMemcpyAsync(x, in_embed, HID_ * sizeof(float), hipMemcpyDeviceToDevice, stream);

    for (int i = 0; i < L_; ++i) {
        ck_rmsnorm<<<1, 256, 0, stream>>>(x, ln1 + (size_t)i * HID_, h, HID_);
        ck_matvec<<<HID_ / 8, 256, 0, stream>>>(Wq + (size_t)i * HID_ * HID_, h, q, nullptr, HID_, HID_);
        ck_matvec<<<HID_ / 8, 256, 0, stream>>>(Wk + (size_t)i * HID_ * HID_, h, k, nullptr, HID_, HID_);
        ck_matvec<<<HID_ / 8, 256, 0, stream>>>(Wv + (size_t)i * HID_ * HID_, h, v, nullptr, HID_, HID_);
        ck_rope_append<<<2, 256, 0, stream>>>(q, k, v, onc, i, posp);

        const float* Kbase = kvc + ((size_t)(2 * i) * KVH_) * (size_t)S_ * D_;
        const float* Vbase = kvc + ((size_t)(2 * i + 1) * KVH_) * (size_t)S_ * D_;
        dim3 sgrid((LEN_ + 15) / 16, H_);
        ck_attn_scores<<<sgrid, 32, 0, stream>>>(Kbase, k, q, sc);
        ck_softmax<<<H_, 256, 0, stream>>>(sc, LEN_);
        ck_attn_av<<<H_, dim3(128, 4), 0, stream>>>(Vbase, v, sc, o);

        ck_matvec<<<HID_ / 8, 256, 0, stream>>>(Wo + (size_t)i * HID_ * HID_, o, x, x, HID_, HID_);
        ck_rmsnorm<<<1, 256, 0, stream>>>(x, ln2 + (size_t)i * HID_, h, HID_);
        ck_gated<<<FF_ / 8, 256, 0, stream>>>(Wg + (size_t)i * FF_ * HID_,
                                              Wu + (size_t)i * FF_ * HID_, h, act, FF_, HID_);
        ck_matvec<<<HID_ / 8, 256, 0, stream>>>(Wd + (size_t)i * HID_ * FF_, act, x, x, HID_, FF_);
    }

    ck_rmsnorm<<<1, 256, 0, stream>>>(x, normw, h, HID_);
    ck_matvec<<<(NCB_ * V_) / 8, 256, 0, stream>>>(lmw, h, logits, nullptr, NCB_ * V_, HID_);
}